// GraphIsomorphismNetwork_13975823581433
// MI455X (gfx1250) — compile-verified
//
#include <hip/hip_runtime.h>
#include <hip/hip_bf16.h>

// ---------------- problem constants (from reference) ----------------
#define GIN_N   100000
#define GIN_E   1600000
#define GIN_FIN 128
#define GIN_H   64
#define GIN_C   16
#define GIN_TILES (GIN_N / 16)   // 6250, exact
#define BN_EPS 1e-5f

typedef __attribute__((ext_vector_type(2))) float v2f;
typedef __attribute__((ext_vector_type(8))) float v8f;

__device__ __forceinline__ v8f wmma_f32_4(v2f a, v2f b, v8f c) {
    // V_WMMA_F32_16X16X4_F32 : D = A(16x4 f32) x B(4x16 f32) + C(16x16 f32)
    return __builtin_amdgcn_wmma_f32_16x16x4_f32(
        /*neg_a=*/false, a, /*neg_b=*/false, b,
        /*c_mod=*/(short)0, c, /*reuse_a=*/false, /*reuse_b=*/false);
}

__device__ __forceinline__ float eluf(float x) {
    return x > 0.0f ? x : (__expf(x) - 1.0f);
}

// ---------------- zero scratch ----------------
__global__ void zero_kernel(float* __restrict__ p, int n) {
    int i = blockIdx.x * blockDim.x + threadIdx.x;
    int st = gridDim.x * blockDim.x;
    for (; i < n; i += st) p[i] = 0.0f;
}

// ---------------- edge scatter: agg[dst] += u[src] (64 ch) ----------------
__global__ __launch_bounds__(256) void scatter_kernel(
        const float* __restrict__ u, const int* __restrict__ src,
        const int* __restrict__ dst, float* __restrict__ agg, int nedges) {
    int idx = blockIdx.x * blockDim.x + threadIdx.x;
    int st  = gridDim.x * blockDim.x;
    int total = nedges * 16;                 // 16 lanes per edge, 4 ch each
    for (int w = idx; w < total; w += st) {
        int e = w >> 4;
        int q = (w & 15) << 2;
        int s = src[e];
        int d = dst[e];
        float4 v = *(const float4*)(u + (size_t)s * GIN_H + q);
        float* a = agg + (size_t)d * GIN_H + q;
        atomicAdd(a + 0, v.x);
        atomicAdd(a + 1, v.y);
        atomicAdd(a + 2, v.z);
        atomicAdd(a + 3, v.w);
    }
}

// ---------------- z = u + agg + b ; accumulate BN sums ----------------
__global__ __launch_bounds__(256) void addsum_kernel(
        const float* __restrict__ u, const float* __restrict__ agg,
        const float* __restrict__ bias, float* __restrict__ z,
        float* __restrict__ bnsum /*[128]*/, int nrows) {
    int c   = threadIdx.x & 63;
    int rpb = blockDim.x >> 6;                       // rows per block per iter
    int r0  = blockIdx.x * rpb + (threadIdx.x >> 6);
    int rst = gridDim.x * rpb;
    float b = bias[c];
    float s = 0.0f, s2 = 0.0f;
    for (int r = r0; r < nrows; r += rst) {
        size_t idx = (size_t)r * GIN_H + c;
        float v = u[idx] + agg[idx] + b;
        z[idx] = v;
        s  += v;
        s2 += v * v;
    }
    atomicAdd(&bnsum[c], s);
    atomicAdd(&bnsum[64 + c], s2);
}

// ---------------- finalize BN: scale = g*rstd, shift = be - mu*scale ----------------
__global__ void bnfinal_kernel(const float* __restrict__ bnsum,
                               const float* __restrict__ g,
                               const float* __restrict__ be,
                               float invN, float* __restrict__ scale,
                               float* __restrict__ shift) {
    int c = threadIdx.x;             // 64 threads
    float mu  = bnsum[c] * invN;
    float var = bnsum[64 + c] * invN - mu * mu;
    float rstd = rsqrtf(var + BN_EPS);
    float sc = g[c] * rstd;
    scale[c] = sc;
    shift[c] = be[c] - mu * sc;
}

// ---------------- WMMA GEMM: D[16r x 64] = op(A[16r x K]) @ W[K x 64] (+epi) ----------------
// TA_BN : A elements transformed as elu(a*scale[k]+shift[k]) on load
// EPI   : output = elu(acc + bias[col]); else raw store
template <int K, bool TA_BN, bool EPI>
__global__ __launch_bounds__(256) void gin_gemm_kernel(
        const float* __restrict__ A, const float* __restrict__ W,
        const float* __restrict__ bias, const float* __restrict__ bnscale,
        const float* __restrict__ bnshift, float* __restrict__ D, int nTiles) {
    __shared__ float sW[K * 64];
    __shared__ float sScale[64];
    __shared__ float sShift[64];

    for (int i = threadIdx.x; i < K * 64; i += blockDim.x) sW[i] = W[i];
    if constexpr (TA_BN) {
        if (threadIdx.x < 64) {
            sScale[threadIdx.x] = bnscale[threadIdx.x];
            sShift[threadIdx.x] = bnshift[threadIdx.x];
        }
    }
    __syncthreads();

    const int lane = threadIdx.x & 31;
    const int m    = lane & 15;            // A row within tile
    const int ko   = (lane >> 4) << 1;     // K-pair offset: 0 or 2
    const int nq   = lane & 15;            // output col within 16-tile
    const int mhi  = (lane >> 4) << 3;     // C/D row offset: 0 or 8

    float bb0 = 0.f, bb1 = 0.f, bb2 = 0.f, bb3 = 0.f;
    if constexpr (EPI) {
        bb0 = bias[nq];      bb1 = bias[16 + nq];
        bb2 = bias[32 + nq]; bb3 = bias[48 + nq];
    }

    int gw  = blockIdx.x * (blockDim.x >> 5) + (threadIdx.x >> 5);
    int gws = gridDim.x * (blockDim.x >> 5);

    for (int tile = gw; tile < nTiles; tile += gws) {
        const float* arow = A + (size_t)(tile * 16 + m) * K;
        v8f c0 = {}, c1 = {}, c2 = {}, c3 = {};
#pragma unroll
        for (int k = 0; k < K; k += 4) {
            v2f a = *(const v2f*)(arow + k + ko);
            if constexpr (TA_BN) {
                a.x = eluf(a.x * sScale[k + ko]     + sShift[k + ko]);
                a.y = eluf(a.y * sScale[k + ko + 1] + sShift[k + ko + 1]);
            }
            const float* w0 = sW + (k + ko) * 64;
            const float* w1 = w0 + 64;
            v2f b;
            b.x = w0[nq];      b.y = w1[nq];      c0 = wmma_f32_4(a, b, c0);
            b.x = w0[16 + nq]; b.y = w1[16 + nq]; c1 = wmma_f32_4(a, b, c1);
            b.x = w0[32 + nq]; b.y = w1[32 + nq]; c2 = wmma_f32_4(a, b, c2);
            b.x = w0[48 + nq]; b.y = w1[48 + nq]; c3 = wmma_f32_4(a, b, c3);
        }
#pragma unroll
        for (int r = 0; r < 8; ++r) {
            int mm = r + mhi;
            float* drow = D + (size_t)(tile * 16 + mm) * 64;
            if constexpr (EPI) {
                drow[nq]      = eluf(c0[r] + bb0);
                drow[16 + nq] = eluf(c1[r] + bb1);
                drow[32 + nq] = eluf(c2[r] + bb2);
                drow[48 + nq] = eluf(c3[r] + bb3);
            } else {
                drow[nq]      = c0[r];
                drow[16 + nq] = c1[r];
                drow[32 + nq] = c2[r];
                drow[48 + nq] = c3[r];
            }
        }
    }
}

// ---------------- head: out = relu([h1|h2]@Wl1 + bl1) @ Wl2 + bl2 ----------------
__global__ __launch_bounds__(256) void head_kernel(
        const float* __restrict__ h1, const float* __restrict__ h2,
        const float* __restrict__ Wl1, const float* __restrict__ bl1,
        const float* __restrict__ Wl2, const float* __restrict__ bl2,
        float* __restrict__ out, int nTiles) {
    __shared__ float sP[8 * 16 * 64];     // per-wave staged p tiles (32KB)
    __shared__ float sW2[64 * 16];        // Wl2 (4KB)
    for (int i = threadIdx.x; i < 64 * 16; i += blockDim.x) sW2[i] = Wl2[i];
    __syncthreads();

    const int lane = threadIdx.x & 31;
    const int wv   = threadIdx.x >> 5;
    const int m    = lane & 15;
    const int ko   = (lane >> 4) << 1;
    const int nq   = lane & 15;
    const int mhi  = (lane >> 4) << 3;
    float* pw = &sP[wv * 16 * 64];

    float bb0 = bl1[nq],      bb1 = bl1[16 + nq];
    float bb2 = bl1[32 + nq], bb3 = bl1[48 + nq];
    float bo  = bl2[nq];

    int gw  = blockIdx.x * (blockDim.x >> 5) + wv;
    int gws = gridDim.x * (blockDim.x >> 5);

    for (int tile = gw; tile < nTiles; tile += gws) {
        const float* a1 = h1 + (size_t)(tile * 16 + m) * 64;
        const float* a2 = h2 + (size_t)(tile * 16 + m) * 64;
        v8f c0 = {}, c1 = {}, c2 = {}, c3 = {};
#pragma unroll
        for (int k = 0; k < 64; k += 4) {           // K-half from h1
            v2f a = *(const v2f*)(a1 + k + ko);
            const float* w0 = Wl1 + (size_t)(k + ko) * 64;
            const float* w1 = w0 + 64;
            v2f b;
            b.x = w0[nq];      b.y = w1[nq];      c0 = wmma_f32_4(a, b, c0);
            b.x = w0[16 + nq]; b.y = w1[16 + nq]; c1 = wmma_f32_4(a, b, c1);
            b.x = w0[32 + nq]; b.y = w1[32 + nq]; c2 = wmma_f32_4(a, b, c2);
            b.x = w0[48 + nq]; b.y = w1[48 + nq]; c3 = wmma_f32_4(a, b, c3);
        }
#pragma unroll
        for (int k = 0; k < 64; k += 4) {           // K-half from h2
            v2f a = *(const v2f*)(a2 + k + ko);
            const float* w0 = Wl1 + (size_t)(64 + k + ko) * 64;
            const float* w1 = w0 + 64;
            v2f b;
            b.x = w0[nq];      b.y = w1[nq];      c0 = wmma_f32_4(a, b, c0);
            b.x = w0[16 + nq]; b.y = w1[16 + nq]; c1 = wmma_f32_4(a, b, c1);
            b.x = w0[32 + nq]; b.y = w1[32 + nq]; c2 = wmma_f32_4(a, b, c2);
            b.x = w0[48 + nq]; b.y = w1[48 + nq]; c3 = wmma_f32_4(a, b, c3);
        }
        // bias + relu, stage p in wave-private LDS (same-wave LDS ops are ordered)
#pragma unroll
        for (int r = 0; r < 8; ++r) {
            int mm = r + mhi;
            pw[mm * 64 + nq]      = fmaxf(c0[r] + bb0, 0.0f);
            pw[mm * 64 + 16 + nq] = fmaxf(c1[r] + bb1, 0.0f);
            pw[mm * 64 + 32 + nq] = fmaxf(c2[r] + bb2, 0.0f);
            pw[mm * 64 + 48 + nq] = fmaxf(c3[r] + bb3, 0.0f);
        }
        // p[16x64] @ Wl2[64x16]
        v8f d = {};
#pragma unroll
        for (int k = 0; k < 64; k += 4) {
            v2f a = *(const v2f*)(pw + m * 64 + k + ko);
            v2f b;
            b.x = sW2[(k + ko) * 16 + nq];
            b.y = sW2[(k + ko + 1) * 16 + nq];
            d = wmma_f32_4(a, b, d);
        }
#pragma unroll
        for (int r = 0; r < 8; ++r) {
            int mm = r + mhi;
            out[(size_t)(tile * 16 + mm) * GIN_C + nq] = d[r] + bo;
        }
    }
}

// ---------------- launcher ----------------
extern "C" void kernel_launch(void* const* d_in, const int* in_sizes, int n_in,
                              void* d_out, int out_size, void* d_ws, size_t ws_size,
                              hipStream_t stream) {
    const float* x   = (const float*)d_in[0];
    const int*   ei  = (const int*)d_in[1];
    const float* W1  = (const float*)d_in[2];
    const float* b1  = (const float*)d_in[3];
    const float* g1  = (const float*)d_in[4];
    const float* be1 = (const float*)d_in[5];
    const float* W2  = (const float*)d_in[6];
    const float* b2  = (const float*)d_in[7];
    const float* W3  = (const float*)d_in[8];
    const float* b3  = (const float*)d_in[9];
    const float* g2  = (const float*)d_in[10];
    const float* be2 = (const float*)d_in[11];
    const float* W4  = (const float*)d_in[12];
    const float* b4  = (const float*)d_in[13];
    const float* Wl1 = (const float*)d_in[14];
    const float* bl1 = (const float*)d_in[15];
    const float* Wl2 = (const float*)d_in[16];
    const float* bl2 = (const float*)d_in[17];

    const int* src = ei;
    const int* dst = ei + GIN_E;

    // workspace layout (floats): u | agg | bnsum(128) | z1/h1 | z2/h2 | scale(64) | shift(64)
    float* ws   = (float*)d_ws;
    size_t NH   = (size_t)GIN_N * GIN_H;
    float* u    = ws;
    float* agg  = u + NH;
    float* bns  = agg + NH;          // 128 floats, contiguous with agg for joint zeroing
    float* z1   = bns + 128;
    float* z2   = z1 + NH;
    float* scl  = z2 + NH;
    float* shf  = scl + 64;

    const float invN = 1.0f / (float)GIN_N;
    const int gemmBlocks    = (GIN_TILES + 7) / 8;   // 8 waves/block, 1 tile/wave
    const int scatterBlocks = 8192;
    const int zeroN         = (int)NH + 128;

    // ---- layer 1 ----
    zero_kernel<<<2048, 256, 0, stream>>>(agg, zeroN);
    gin_gemm_kernel<GIN_FIN, false, false><<<gemmBlocks, 256, 0, stream>>>(
        x, W1, nullptr, nullptr, nullptr, u, GIN_TILES);
    scatter_kernel<<<scatterBlocks, 256, 0, stream>>>(u, src, dst, agg, GIN_E);
    addsum_kernel<<<1024, 256, 0, stream>>>(u, agg, b1, z1, bns, GIN_N);
    bnfinal_kernel<<<1, 64, 0, stream>>>(bns, g1, be1, invN, scl, shf);
    gin_gemm_kernel<GIN_H, true, true><<<gemmBlocks, 256, 0, stream>>>(
        z1, W2, b2, scl, shf, z1, GIN_TILES);            // h1 in-place

    // ---- layer 2 ----
    zero_kernel<<<2048, 256, 0, stream>>>(agg, zeroN);
    gin_gemm_kernel<GIN_H, false, false><<<gemmBlocks, 256, 0, stream>>>(
        z1, W3, nullptr, nullptr, nullptr, u, GIN_TILES);
    scatter_kernel<<<scatterBlocks, 256, 0, stream>>>(u, src, dst, agg, GIN_E);
    addsum_kernel<<<1024, 256, 0, stream>>>(u, agg, b3, z2, bns, GIN_N);
    bnfinal_kernel<<<1, 64, 0, stream>>>(bns, g2, be2, invN, scl, shf);
    gin_gemm_kernel<GIN_H, true, true><<<gemmBlocks, 256, 0, stream>>>(
        z2, W4, b4, scl, shf, z2, GIN_TILES);            // h2 in-place

    // ---- head ----
    head_kernel<<<gemmBlocks, 256, 0, stream>>>(
        z1, z2, Wl1, bl1, Wl2, bl2, (float*)d_out, GIN_TILES);
}